// MomentumAdvection2D_67147518706038
// MI455X (gfx1250) — compile-verified
//
#include <hip/hip_runtime.h>

// ---------------------------------------------------------------------------
// MomentumAdvection2D (Arakawa C-grid, Sadourny E-scheme) for MI455X/gfx1250.
// HBM-bound stencil (~2 FLOP/byte): stage u/v tiles to LDS with gfx1250
// async-to-LDS DMA (b128 payloads, ASYNCcnt), compute from LDS, stream the
// outputs with non-temporal b128 stores.
// ---------------------------------------------------------------------------

namespace {
constexpr int NYg = 4096;
constexpr int NXg = 4096;
constexpr int TX  = 128;          // output tile width
constexpr int TY  = 32;           // output tile height
constexpr int LH  = TY + 2;       // 34 staged rows (1-wide halo)
constexpr int LSTR = 136;         // LDS row stride in floats (544 B, 16B-aligned)
// LDS column layout: col 3 = left halo (gi = gx0-1), cols 4..131 = main
// (gi = gx0..gx0+127, 16B-aligned quads), col 132 = right halo (gi = gx0+128).
constexpr int NQUAD = LH * (TX / 4);  // 1088 main float4 quads per array
constexpr int NHALO = LH * 2;         // 68 halo scalars per array
constexpr int BLK = 512;              // 16 waves (wave32)
}

typedef float v4f __attribute__((ext_vector_type(4)));
typedef int   v4i __attribute__((ext_vector_type(4)));

#define GLOBAL_AS __attribute__((address_space(1)))
#define LDS_AS    __attribute__((address_space(3)))

// Select async-to-LDS path: builtin > inline asm > plain copy.
#if defined(__gfx1250__)
  #if __has_builtin(__builtin_amdgcn_global_load_async_to_lds_b32)
    #define ASYNC_MODE 1
  #else
    #define ASYNC_MODE 2
  #endif
#else
  #define ASYNC_MODE 0
#endif

#if ASYNC_MODE == 1 && __has_builtin(__builtin_amdgcn_global_load_async_to_lds_b128)
  #define HAVE_ASYNC_B128 1
#else
  #define HAVE_ASYNC_B128 0
#endif

__device__ __forceinline__ GLOBAL_AS int* to_global_i(const void* p) {
    // Flat pointers to global memory share the 64-bit VA representation.
    return (GLOBAL_AS int*)(unsigned long long)p;
}
__device__ __forceinline__ LDS_AS int* to_lds_i(void* p) {
    // ISA 10.2: flat LDS address low 32 bits == LDS byte offset.
    return (LDS_AS int*)(unsigned)(unsigned long long)p;
}

__device__ __forceinline__ void async_copy_f32(const float* gsrc, float* ldst) {
#if ASYNC_MODE == 1
    __builtin_amdgcn_global_load_async_to_lds_b32(to_global_i(gsrc), to_lds_i(ldst), 0, 0);
#elif ASYNC_MODE == 2
    unsigned lds_off = (unsigned)(unsigned long long)ldst;   // low 32 = LDS offset
    unsigned long long gaddr = (unsigned long long)gsrc;
    asm volatile("global_load_async_to_lds_b32 %0, %1, off"
                 :: "v"(lds_off), "v"(gaddr) : "memory");
#else
    *ldst = *gsrc;
#endif
}

__device__ __forceinline__ void async_copy_f32x4(const float* gsrc, float* ldst) {
#if HAVE_ASYNC_B128
    __builtin_amdgcn_global_load_async_to_lds_b128(
        (GLOBAL_AS v4i*)(unsigned long long)gsrc,
        (LDS_AS v4i*)(unsigned)(unsigned long long)ldst, 0, 0);
#elif ASYNC_MODE == 1
    async_copy_f32(gsrc + 0, ldst + 0);
    async_copy_f32(gsrc + 1, ldst + 1);
    async_copy_f32(gsrc + 2, ldst + 2);
    async_copy_f32(gsrc + 3, ldst + 3);
#elif ASYNC_MODE == 2
    unsigned lds_off = (unsigned)(unsigned long long)ldst;
    unsigned long long gaddr = (unsigned long long)gsrc;
    asm volatile("global_load_async_to_lds_b128 %0, %1, off"
                 :: "v"(lds_off), "v"(gaddr) : "memory");
#else
    ldst[0] = gsrc[0]; ldst[1] = gsrc[1]; ldst[2] = gsrc[2]; ldst[3] = gsrc[3];
#endif
}

__device__ __forceinline__ void async_wait_all() {
#if ASYNC_MODE == 1
  #if __has_builtin(__builtin_amdgcn_s_wait_asynccnt)
    __builtin_amdgcn_s_wait_asynccnt(0);
  #else
    asm volatile("s_wait_asynccnt 0" ::: "memory");
  #endif
#elif ASYNC_MODE == 2
    asm volatile("s_wait_asynccnt 0" ::: "memory");
#endif
}

__global__ __launch_bounds__(BLK, 2)
void MomentumAdvection2D_67147518706038_kernel(const float* __restrict__ u,
                                               const float* __restrict__ v,
                                               float* __restrict__ du,
                                               float* __restrict__ dv)
{
    __shared__ float su[LH * LSTR];
    __shared__ float sv[LH * LSTR];

    const int tid = threadIdx.x;
    const int gx0 = blockIdx.x * TX;
    const int gy0 = blockIdx.y * TY;

    // ---- Stage main tile body: b128 async DMA, 16B-aligned both sides. ----
    // Row r (0..33) <-> global row clamp(gy0-1+r); quad m (0..31) <-> global
    // cols gx0+4m..gx0+4m+3 -> LDS cols 4+4m..7+4m. Clamped/duplicated rows
    // only feed ring outputs, which are masked to zero below.
#pragma unroll
    for (int k = 0; k < (NQUAD + BLK - 1) / BLK; ++k) {
        const int flat = k * BLK + tid;
        if (flat < NQUAD) {
            const int r = flat >> 5;          // staged row 0..33
            const int m = flat & 31;          // quad index 0..31
            int gj = gy0 - 1 + r; gj = gj < 0 ? 0 : (gj > NYg - 1 ? NYg - 1 : gj);
            const size_t goff = (size_t)gj * NXg + (size_t)(gx0 + m * 4);
            const int loff = r * LSTR + 4 + m * 4;
            async_copy_f32x4(u + goff, &su[loff]);
            async_copy_f32x4(v + goff, &sv[loff]);
        }
    }

    // ---- Stage halo columns (LDS cols 3 and 132): b32 async DMA. ----
    if (tid < NHALO) {
        const int r = tid >> 1;
        const int side = tid & 1;
        int gj = gy0 - 1 + r; gj = gj < 0 ? 0 : (gj > NYg - 1 ? NYg - 1 : gj);
        int gi = side ? (gx0 + TX) : (gx0 - 1);
        gi = gi < 0 ? 0 : (gi > NXg - 1 ? NXg - 1 : gi);
        const size_t goff = (size_t)gj * NXg + gi;
        const int loff = r * LSTR + (side ? 132 : 3);
        async_copy_f32(u + goff, &su[loff]);
        async_copy_f32(v + goff, &sv[loff]);
    }

    async_wait_all();
    __syncthreads();

    // ---- Compute: each thread does 2 rows x 4 cols; b128 NT stores. ----
    const int qx = tid & 31;   // quad-column 0..31  -> cols qx*4 .. qx*4+3
    const int ry = tid >> 5;   // 0..15              -> rows ry, ry+16

#pragma unroll
    for (int h = 0; h < 2; ++h) {
        const int ly = ry + h * 16;
        const int j  = gy0 + ly;
        const int R  = ly + 1;
        v4f duq, dvq;
#pragma unroll
        for (int q = 0; q < 4; ++q) {
            const int lx = qx * 4 + q;
            const int i  = gx0 + lx;
            const int L  = lx + 4;            // main cols start at LDS col 4
            const float* uR0 = &su[(R - 1) * LSTR];
            const float* uR1 = &su[(R    ) * LSTR];
            const float* uR2 = &su[(R + 1) * LSTR];
            const float* vR0 = &sv[(R - 1) * LSTR];
            const float* vR1 = &sv[(R    ) * LSTR];
            const float* vR2 = &sv[(R + 1) * LSTR];

            const float uC = uR1[L], uW = uR1[L - 1], uE = uR1[L + 1];
            const float uN = uR0[L], uS = uR2[L],     uSW = uR2[L - 1];
            const float vC = vR1[L], vW = vR1[L - 1], vE = vR1[L + 1];
            const float vN = vR0[L], vNE = vR0[L + 1], vS = vR2[L];

            // zeta at (j,i), (j-1,i), (j,i-1)
            const float z  = (vE  - vC) - (uS - uC);
            const float zN = (vNE - vN) - (uC - uN);
            const float zW = (vC  - vW) - (uSW - uW);

            // kinetic energy at T-points (j,i), (j,i+1), (j+1,i)
            const float uT  = 0.5f * (uC + uW),  vT  = 0.5f * (vC + vN);
            const float uTE = 0.5f * (uE + uC),  vTE = 0.5f * (vE + vNE);
            const float uTS = 0.5f * (uS + uSW), vTS = 0.5f * (vS + vC);
            const float K   = 0.5f * (uT  * uT  + vT  * vT);
            const float KE  = 0.5f * (uTE * uTE + vTE * vTE);
            const float KS  = 0.5f * (uTS * uTS + vTS * vTS);

            const float vU = 0.25f * (vC + vE + vN + vNE);   // V_to_U
            const float uV = 0.25f * (uC + uW + uS + uSW);   // U_to_V

            const float duv =  0.5f * (z + zN) * vU - (KE - K);
            const float dvv = -0.5f * (z + zW) * uV - (KS - K);

            const bool valid = (j >= 2) && (j <= NYg - 3) && (i >= 2) && (i <= NXg - 3);
            duq[q] = valid ? duv : 0.0f;
            dvq[q] = valid ? dvv : 0.0f;
        }
        const size_t ooff = (size_t)j * NXg + (size_t)(gx0 + qx * 4);
        __builtin_nontemporal_store(duq, (v4f*)(du + ooff));
        __builtin_nontemporal_store(dvq, (v4f*)(dv + ooff));
    }
}

extern "C" void kernel_launch(void* const* d_in, const int* in_sizes, int n_in,
                              void* d_out, int out_size, void* d_ws, size_t ws_size,
                              hipStream_t stream) {
    (void)in_sizes; (void)n_in; (void)out_size; (void)d_ws; (void)ws_size;
    const float* u = (const float*)d_in[0];
    const float* v = (const float*)d_in[1];
    float* du = (float*)d_out;
    float* dv = du + (size_t)NYg * NXg;

    dim3 grid(NXg / TX, NYg / TY);   // 32 x 128 tiles
    dim3 block(BLK);
    MomentumAdvection2D_67147518706038_kernel<<<grid, block, 0, stream>>>(u, v, du, dv);
}